// MSRG_66709432042123
// MI455X (gfx1250) — compile-verified
//
#include <hip/hip_runtime.h>

typedef __attribute__((ext_vector_type(2))) float v2f;
typedef __attribute__((ext_vector_type(8))) float v8f;

#define WMMA_F32(a, b, c) \
  __builtin_amdgcn_wmma_f32_16x16x4_f32(false, (a), false, (b), (short)0, (c), false, false)

__device__ __forceinline__ float sigf(float x) { return 1.f / (1.f + expf(-x)); }

// ------------------------------------------------------------------
// Embedding gather: enc[i, 0:300] = emb[sentences[i], 0:300]
// ------------------------------------------------------------------
__global__ void k_gather(const int* __restrict__ idx, const float* __restrict__ emb,
                         float* __restrict__ enc)
{
  long i = blockIdx.x;
  long tok = idx[i];
  const float* s = emb + tok * 300;
  float* d = enc + i * 300;
  for (int c = threadIdx.x; c < 300; c += blockDim.x) d[c] = s[c];
}

// ------------------------------------------------------------------
// Generic fp32 WMMA GEMM: C = alpha*(A @ op(B)) + bias (+ C if beta)
// One 16x16 tile per wave, 4 waves per block, batched via gridDim.z.
// A: [M,K] lda ; B: [K,N] ldb (or [N,K] if TRANSB) ; C: [M,N] ldc.
// M % 64 == 0, N % 16 == 0 at every call site. Ragged K handled by a
// single guarded tail chunk AFTER the branch-free main loop.
// ------------------------------------------------------------------
template <bool TRANSB>
__global__ void __launch_bounds__(128)
k_gemm(const float* __restrict__ A, const float* __restrict__ B,
       const float* __restrict__ bias, float* __restrict__ C,
       int M, int N, int K, int lda, int ldb, int ldc,
       long long sA, long long sB, long long sC,
       float alpha, int beta)
{
  A += (long long)blockIdx.z * sA;
  B += (long long)blockIdx.z * sB;
  C += (long long)blockIdx.z * sC;
  const int lane  = threadIdx.x & 31;
  const int wave  = threadIdx.x >> 5;
  const int tm    = blockIdx.y * 4 + wave;
  const int tn    = blockIdx.x;
  const int row   = tm * 16 + (lane & 15);
  const int col   = tn * 16 + (lane & 15);
  const int khalf = (lane >> 4) * 2;

  // per-lane base pointers (khalf folded in)
  const float* Ap = A + (long)row * lda + khalf;
  const float* Bp = TRANSB ? (B + (long)col * ldb + khalf)   // b.x/b.y contiguous
                           : (B + (long)khalf * ldb + col);  // b.y one ldb below b.x

  v8f acc = {0.f, 0.f, 0.f, 0.f, 0.f, 0.f, 0.f, 0.f};
  const int Kmain = K & ~3;
  for (int k = 0; k < Kmain; k += 4) {
    v2f a, b;
    a.x = Ap[k];
    a.y = Ap[k + 1];
    if (TRANSB) {
      b.x = Bp[k];
      b.y = Bp[k + 1];
    } else {
      b.x = Bp[(long)k * ldb];
      b.y = Bp[(long)k * ldb + ldb];
    }
    acc = WMMA_F32(a, b, acc);
  }
  if (K & 3) {  // single guarded tail chunk (K = 74/35 cases)
    const int kk = Kmain + khalf;
    v2f a = {0.f, 0.f}, b = {0.f, 0.f};
    if (kk < K) {
      a.x = Ap[Kmain];
      b.x = TRANSB ? Bp[Kmain] : Bp[(long)Kmain * ldb];
    }
    if (kk + 1 < K) {
      a.y = Ap[Kmain + 1];
      b.y = TRANSB ? Bp[Kmain + 1] : Bp[(long)Kmain * ldb + ldb];
    }
    acc = WMMA_F32(a, b, acc);
  }

  const float bv = bias ? bias[col] : 0.f;
  const int mbase = tm * 16 + ((lane >> 4) ? 8 : 0);
  for (int r = 0; r < 8; ++r) {
    long o = (long)(mbase + r) * ldc + col;
    float v = alpha * acc[r] + bv;
    if (beta) v += C[o];
    C[o] = v;
  }
}

// ------------------------------------------------------------------
// Row softmax over 256-wide rows, in place. grid = #rows, block = 256.
// ------------------------------------------------------------------
__global__ void k_softmax256(float* __restrict__ s)
{
  __shared__ float red[256];
  float* p = s + (long)blockIdx.x * 256;
  const int t = threadIdx.x;
  float v = p[t];
  red[t] = v; __syncthreads();
  for (int o = 128; o > 0; o >>= 1) {
    if (t < o) red[t] = fmaxf(red[t], red[t + o]);
    __syncthreads();
  }
  const float mx = red[0];
  __syncthreads();
  const float e = expf(v - mx);
  red[t] = e; __syncthreads();
  for (int o = 128; o > 0; o >>= 1) {
    if (t < o) red[t] += red[t + o];
    __syncthreads();
  }
  p[t] = e / red[0];
}

// ------------------------------------------------------------------
// Persistent biGRU direction scan. H=64, B=64, S=256.
// 16 waves: wave = (m-tile[0..3], j-tile[0..3]); each wave owns all 3
// gate columns for its (m,j) block -> cell update is register local.
// h lives in LDS (padded stride 65); h@Wh via fp32 WMMA from LDS.
// xg: [64,256,192] precomputed x@Wi+bi. y gets +col-offset, stride ldy.
// ------------------------------------------------------------------
__global__ void __launch_bounds__(512)
k_gru_scan(const float* __restrict__ xg, const float* __restrict__ Wh,
           const float* __restrict__ bh, float* __restrict__ y,
           int ldy, int reverse)
{
  __shared__ float h_s[64 * 65];
  const int lane = threadIdx.x & 31;
  const int wave = threadIdx.x >> 5;   // 0..15
  const int tmi  = wave >> 2;          // m tile 0..3
  const int tji  = wave & 3;           // j tile 0..3
  const int l15  = lane & 15;
  const int hi   = lane >> 4;
  const int arow = tmi * 16 + l15;
  const int jc   = tji * 16 + l15;
  const int khalf = hi * 2;

  for (int i = threadIdx.x; i < 64 * 65; i += blockDim.x) h_s[i] = 0.f;
  v8f hreg = {0.f, 0.f, 0.f, 0.f, 0.f, 0.f, 0.f, 0.f};
  __syncthreads();

  const float bhr = bh[jc], bhz = bh[64 + jc], bhn = bh[128 + jc];
  for (int step = 0; step < 256; ++step) {
    const int t = reverse ? 255 - step : step;
    v8f ar = {0.f,0.f,0.f,0.f,0.f,0.f,0.f,0.f};
    v8f az = ar, an = ar;
    for (int k = 0; k < 64; k += 4) {
      const int kk = k + khalf;
      v2f a, b;
      a.x = h_s[arow * 65 + kk];
      a.y = h_s[arow * 65 + kk + 1];
      b.x = Wh[kk * 192 + jc];        b.y = Wh[(kk + 1) * 192 + jc];
      ar = WMMA_F32(a, b, ar);
      b.x = Wh[kk * 192 + 64 + jc];   b.y = Wh[(kk + 1) * 192 + 64 + jc];
      az = WMMA_F32(a, b, az);
      b.x = Wh[kk * 192 + 128 + jc];  b.y = Wh[(kk + 1) * 192 + 128 + jc];
      an = WMMA_F32(a, b, an);
    }
    __syncthreads();  // everyone done reading previous h
    for (int r = 0; r < 8; ++r) {
      const int m = tmi * 16 + hi * 8 + r;
      const float* xr = xg + ((long)m * 256 + t) * 192;
      float rg = sigf(xr[jc]       + ar[r] + bhr);
      float zg = sigf(xr[64 + jc]  + az[r] + bhz);
      float ng = tanhf(xr[128 + jc] + rg * (an[r] + bhn));
      float hn = (1.f - zg) * ng + zg * hreg[r];
      hreg[r] = hn;
      h_s[m * 65 + jc] = hn;
      y[((long)m * 256 + t) * ldy + jc] = hn;
    }
    __syncthreads();
  }
}

// ------------------------------------------------------------------
// Persistent biLSTM direction scan. H=128, 4 gates (i,f,g,o).
// 32 waves: (m-tile[0..3], j-tile[0..7]); c state stays in registers.
// xg: [64,256,512]; Wh: [128,512]; y + col-offset, stride ldy.
// ------------------------------------------------------------------
__global__ void __launch_bounds__(1024)
k_lstm_scan(const float* __restrict__ xg, const float* __restrict__ Wh,
            const float* __restrict__ bh, float* __restrict__ y,
            int ldy, int reverse)
{
  __shared__ float h_s[64 * 129];
  const int lane = threadIdx.x & 31;
  const int wave = threadIdx.x >> 5;   // 0..31
  const int tmi  = wave >> 3;          // 0..3
  const int tji  = wave & 7;           // 0..7
  const int l15  = lane & 15;
  const int hi   = lane >> 4;
  const int arow = tmi * 16 + l15;
  const int jc   = tji * 16 + l15;
  const int khalf = hi * 2;

  for (int i = threadIdx.x; i < 64 * 129; i += blockDim.x) h_s[i] = 0.f;
  v8f creg = {0.f, 0.f, 0.f, 0.f, 0.f, 0.f, 0.f, 0.f};
  __syncthreads();

  const float bi_ = bh[jc], bf_ = bh[128 + jc], bg_ = bh[256 + jc], bo_ = bh[384 + jc];
  for (int step = 0; step < 256; ++step) {
    const int t = reverse ? 255 - step : step;
    v8f gi = {0.f,0.f,0.f,0.f,0.f,0.f,0.f,0.f};
    v8f gf = gi, gg = gi, go = gi;
    for (int k = 0; k < 128; k += 4) {
      const int kk = k + khalf;
      v2f a, b;
      a.x = h_s[arow * 129 + kk];
      a.y = h_s[arow * 129 + kk + 1];
      b.x = Wh[kk * 512 + jc];        b.y = Wh[(kk + 1) * 512 + jc];
      gi = WMMA_F32(a, b, gi);
      b.x = Wh[kk * 512 + 128 + jc];  b.y = Wh[(kk + 1) * 512 + 128 + jc];
      gf = WMMA_F32(a, b, gf);
      b.x = Wh[kk * 512 + 256 + jc];  b.y = Wh[(kk + 1) * 512 + 256 + jc];
      gg = WMMA_F32(a, b, gg);
      b.x = Wh[kk * 512 + 384 + jc];  b.y = Wh[(kk + 1) * 512 + 384 + jc];
      go = WMMA_F32(a, b, go);
    }
    __syncthreads();
    for (int r = 0; r < 8; ++r) {
      const int m = tmi * 16 + hi * 8 + r;
      const float* xr = xg + ((long)m * 256 + t) * 512;
      float iv = sigf(xr[jc]        + gi[r] + bi_);
      float fv = sigf(xr[128 + jc]  + gf[r] + bf_);
      float gv = tanhf(xr[256 + jc] + gg[r] + bg_);
      float ov = sigf(xr[384 + jc]  + go[r] + bo_);
      float c  = fv * creg[r] + iv * gv;
      float h  = ov * tanhf(c);
      creg[r] = c;
      h_s[m * 129 + jc] = h;
      y[((long)m * 256 + t) * ldy + jc] = h;
    }
    __syncthreads();
  }
}

// ------------------------------------------------------------------
// Persistent RGN scan (GRU-style cell, no recurrent bias), H=128,
// input dim 256 pre-projected to xg [64,256,384]. Only final h out.
// ------------------------------------------------------------------
__global__ void __launch_bounds__(1024)
k_rgn_scan(const float* __restrict__ xg, const float* __restrict__ Wh,
           const float* __restrict__ h0, float* __restrict__ hout,
           int reverse)
{
  __shared__ float h_s[64 * 129];
  const int lane = threadIdx.x & 31;
  const int wave = threadIdx.x >> 5;   // 0..31
  const int tmi  = wave >> 3;
  const int tji  = wave & 7;
  const int l15  = lane & 15;
  const int hi   = lane >> 4;
  const int arow = tmi * 16 + l15;
  const int jc   = tji * 16 + l15;
  const int khalf = hi * 2;

  v8f hreg;
  for (int r = 0; r < 8; ++r) {
    const int m = tmi * 16 + hi * 8 + r;
    float h = h0[m * 128 + jc];
    hreg[r] = h;
    h_s[m * 129 + jc] = h;
  }
  __syncthreads();

  for (int step = 0; step < 256; ++step) {
    const int t = reverse ? 255 - step : step;
    v8f ar = {0.f,0.f,0.f,0.f,0.f,0.f,0.f,0.f};
    v8f az = ar, an = ar;
    for (int k = 0; k < 128; k += 4) {
      const int kk = k + khalf;
      v2f a, b;
      a.x = h_s[arow * 129 + kk];
      a.y = h_s[arow * 129 + kk + 1];
      b.x = Wh[kk * 384 + jc];        b.y = Wh[(kk + 1) * 384 + jc];
      ar = WMMA_F32(a, b, ar);
      b.x = Wh[kk * 384 + 128 + jc];  b.y = Wh[(kk + 1) * 384 + 128 + jc];
      az = WMMA_F32(a, b, az);
      b.x = Wh[kk * 384 + 256 + jc];  b.y = Wh[(kk + 1) * 384 + 256 + jc];
      an = WMMA_F32(a, b, an);
    }
    __syncthreads();
    for (int r = 0; r < 8; ++r) {
      const int m = tmi * 16 + hi * 8 + r;
      const float* xr = xg + ((long)m * 256 + t) * 384;
      float rg = sigf(xr[jc]        + ar[r]);
      float zg = sigf(xr[128 + jc]  + az[r]);
      float ng = tanhf(xr[256 + jc] + rg * an[r]);
      float hn = (1.f - zg) * ng + zg * hreg[r];
      hreg[r] = hn;
      h_s[m * 129 + jc] = hn;
    }
    __syncthreads();
  }
  for (int r = 0; r < 8; ++r) {
    const int m = tmi * 16 + hi * 8 + r;
    hout[m * 128 + jc] = hreg[r];
  }
}

// ------------------------------------------------------------------
// Head: out[b] = (leaky_relu((o0+o1) @ W1 + b1)) @ W2 + b2
// ------------------------------------------------------------------
__global__ void k_fc(const float* __restrict__ o0, const float* __restrict__ o1,
                     const float* __restrict__ W1, const float* __restrict__ b1,
                     const float* __restrict__ W2, const float* __restrict__ b2,
                     float* __restrict__ out)
{
  const int b = threadIdx.x;
  if (b >= 64) return;
  float acc2 = b2[0];
  for (int u = 0; u < 32; ++u) {
    float a = b1[u];
    for (int j = 0; j < 128; ++j)
      a += (o0[b * 128 + j] + o1[b * 128 + j]) * W1[j * 32 + u];
    a = a > 0.f ? a : 0.01f * a;   // leaky_relu, slope 0.01
    acc2 += a * W2[u];
  }
  out[b] = acc2;
}

// ------------------------------------------------------------------
extern "C" void kernel_launch(void* const* d_in, const int* in_sizes, int n_in,
                              void* d_out, int out_size, void* d_ws, size_t ws_size,
                              hipStream_t stream)
{
  (void)in_sizes; (void)n_in; (void)out_size; (void)ws_size;

  const int*   sentences = (const int*)d_in[0];
  const float* acoustic  = (const float*)d_in[1];
  const float* video     = (const float*)d_in[2];
  const float* state0    = (const float*)d_in[3];
  const float* state1    = (const float*)d_in[4];
  const float* emb       = (const float*)d_in[5];
  const float* proj_W[3] = {(const float*)d_in[6], (const float*)d_in[8], (const float*)d_in[10]};
  const float* proj_b[3] = {(const float*)d_in[7], (const float*)d_in[9], (const float*)d_in[11]};
  const float* gru_Wi  = (const float*)d_in[12];
  const float* gru_Wh  = (const float*)d_in[13];
  const float* gru_bi  = (const float*)d_in[14];
  const float* gru_bh  = (const float*)d_in[15];
  const float* lstm_Wi = (const float*)d_in[16];
  const float* lstm_Wh = (const float*)d_in[17];
  const float* lstm_bi = (const float*)d_in[18];
  const float* lstm_bh = (const float*)d_in[19];
  const float* rgn_Wx  = (const float*)d_in[20];
  const float* rgn_Wh  = (const float*)d_in[21];
  const float* rgn_b   = (const float*)d_in[22];
  const float* fc1_W   = (const float*)d_in[23];
  const float* fc1_b   = (const float*)d_in[24];
  const float* fc2_W   = (const float*)d_in[25];
  const float* fc2_b   = (const float*)d_in[26];

  // Workspace layout (floats); total ~146 MB with aliasing.
  float* ws    = (float*)d_ws;
  float* enc   = ws;                   // 16384 x 300 ; later reused as step_fusion 16384 x 256
  float* X     = enc + 4915200;        // 16384 x 384 : [x_l | x_a | x_v]
  float* tstep = X + 6291456;          // 16384 x 384 : [fin_txt | fin_aud | fin_vis]
  float* seq   = tstep + 6291456;      // 16384 x 128 (reused per modality)
  float* xgA   = seq + 2097152;        // up to 16384 x 512 ; also attention scores 64x256x256
  float* xgB   = xgA + 8388608;        // up to 16384 x 512
  float* out0  = xgB + 8388608;        // 64 x 128
  float* out1  = out0 + 8192;          // 64 x 128

  auto gemm = [&](const float* A, const float* B, const float* bias, float* C,
                  int M, int N, int K, int lda, int ldb, int ldc,
                  long long sA, long long sB, long long sC, int nb,
                  float alpha, int transB, int beta) {
    dim3 g(N / 16, M / 64, nb);
    if (transB)
      k_gemm<true><<<g, 128, 0, stream>>>(A, B, bias, C, M, N, K, lda, ldb, ldc,
                                          sA, sB, sC, alpha, beta);
    else
      k_gemm<false><<<g, 128, 0, stream>>>(A, B, bias, C, M, N, K, lda, ldb, ldc,
                                           sA, sB, sC, alpha, beta);
  };

  // 1) embedding gather
  k_gather<<<16384, 256, 0, stream>>>(sentences, emb, enc);

  // 2) per-modality: projection -> biGRU (input GEMM + persistent scan)
  const float* modx[3] = {enc, acoustic, video};
  const int    modk[3] = {300, 74, 35};
  for (int m = 0; m < 3; ++m) {
    gemm(modx[m], proj_W[m], proj_b[m], seq,
         16384, 128, modk[m], modk[m], 128, 128, 0, 0, 0, 1, 1.f, 0, 0);
    for (int d = 0; d < 2; ++d) {
      float* xg = d ? xgB : xgA;
      const int w = m * 2 + d;
      gemm(seq, gru_Wi + (long)w * 128 * 192, gru_bi + w * 192, xg,
           16384, 192, 128, 128, 192, 192, 0, 0, 0, 1, 1.f, 0, 0);
      k_gru_scan<<<1, 512, 0, stream>>>(xg, gru_Wh + (long)w * 64 * 192,
                                        gru_bh + w * 192,
                                        X + m * 128 + d * 64, 384, d);
    }
  }

  // 3) JCAF cross-attention: tstep = X, then += softmax(q kv^T / sqrt(128)) kv
  hipMemcpyAsync(tstep, X, (size_t)16384 * 384 * sizeof(float),
                 hipMemcpyDeviceToDevice, stream);
  float* scores = xgA;                          // alias: GRU xg no longer needed
  const float scale = 0.088388347648318447f;    // 1/sqrt(128)
  const int prs[6][2] = {{0,1},{0,2},{1,0},{1,2},{2,0},{2,1}};
  for (int p = 0; p < 6; ++p) {
    const int qi = prs[p][0], kj = prs[p][1];
    gemm(X + qi * 128, X + kj * 128, nullptr, scores,
         256, 256, 128, 384, 384, 256, 98304, 98304, 65536, 64, scale, 1, 0);
    k_softmax256<<<16384, 256, 0, stream>>>(scores);
    gemm(scores, X + kj * 128, nullptr, tstep + qi * 128,
         256, 128, 256, 256, 384, 384, 65536, 98304, 98304, 64, 1.f, 0, 1);
  }

  // 4) biLSTM over tstep -> step_fusion (reuses enc buffer)
  float* sf = enc;
  for (int d = 0; d < 2; ++d) {
    float* xg = d ? xgB : xgA;
    gemm(tstep, lstm_Wi + (long)d * 384 * 512, lstm_bi + d * 512, xg,
         16384, 512, 384, 384, 512, 512, 0, 0, 0, 1, 1.f, 0, 0);
    k_lstm_scan<<<1, 1024, 0, stream>>>(xg, lstm_Wh + (long)d * 128 * 512,
                                        lstm_bh + d * 512, sf + d * 128, 256, d);
  }

  // 5) RGN scans (forward + reversed), only final states kept
  for (int d = 0; d < 2; ++d) {
    float* xg = d ? xgB : xgA;
    gemm(sf, rgn_Wx + (long)d * 256 * 384, rgn_b + d * 384, xg,
         16384, 384, 256, 256, 384, 384, 0, 0, 0, 1, 1.f, 0, 0);
    k_rgn_scan<<<1, 1024, 0, stream>>>(xg, rgn_Wh + (long)d * 128 * 384,
                                       d ? state1 : state0, d ? out1 : out0, d);
  }

  // 6) MLP head
  k_fc<<<1, 64, 0, stream>>>(out0, out1, fc1_W, fc1_b, fc2_W, fc2_b, (float*)d_out);
}